// DeformableResidualBlock_24300924961373
// MI455X (gfx1250) — compile-verified
//
#include <hip/hip_runtime.h>

typedef float v2f __attribute__((ext_vector_type(2)));
typedef float v8f __attribute__((ext_vector_type(8)));

constexpr int Bc = 4, Cc = 64, Hc = 128, Wc = 128;
constexpr int HW   = Hc * Wc;      // 16384 = 1<<14
constexpr int Kdim = Cc * 9;       // 576
constexpr float NEGS = 0.01f;

__device__ __forceinline__ int iclamp(int v, int lo, int hi) {
    return v < lo ? lo : (v > hi ? hi : v);
}

__device__ __forceinline__ v8f wmma4(v2f A, v2f B, v8f C) {
    return __builtin_amdgcn_wmma_f32_16x16x4_f32(
        false, A, false, B, (short)0, C, false, false);
}

// ---------------------------------------------------------------------------
// Pad the 18x576 offset-conv weights to 32x576 (rows 18..31 zero).
// ---------------------------------------------------------------------------
__global__ __launch_bounds__(256)
void padw_kernel(const float* __restrict__ woff, float* __restrict__ wpad)
{
    int i = blockIdx.x * 256 + threadIdx.x;          // < 32*576
    if (i < 32 * Kdim)
        wpad[i] = (i < 18 * Kdim) ? woff[i] : 0.f;
}

// ---------------------------------------------------------------------------
// Offset conv as WMMA GEMM: M=16 px / wave, N=32 (store 18), K=576.
// ---------------------------------------------------------------------------
__global__ __launch_bounds__(128)
void offconv_kernel(const float* __restrict__ x,
                    const float* __restrict__ wpad,   // (32,576)
                    const float* __restrict__ boff,
                    float* __restrict__ off)
{
    __shared__ __align__(16) int   s_idx[4 * 144];
    __shared__ __align__(16) float s_msk[4 * 144];

    const int wv   = threadIdx.x >> 5;
    const int lane = threadIdx.x & 31;
    const int tile = blockIdx.x * 4 + wv;
    const int p0 = tile << 4;
    const int b  = p0 >> 14;
    const int hw = p0 & (HW - 1);
    const int h0 = hw >> 7;
    const int w0 = hw & (Wc - 1);

    for (int ent = lane; ent < 144; ent += 32) {
        int i = ent / 9, t = ent - i * 9;
        int ty = t / 3, tx = t - ty * 3;
        int yy = h0 + ty - 1;
        int xx = w0 + i + tx - 1;
        bool valid = (yy >= 0) && (yy < Hc) && (xx >= 0) && (xx < Wc);
        s_idx[wv * 144 + ent] = iclamp(yy, 0, Hc - 1) * Wc + iclamp(xx, 0, Wc - 1);
        s_msk[wv * 144 + ent] = valid ? 1.f : 0.f;
    }
    __syncthreads();

    const int hl   = lane >> 4;
    const int koff = hl << 1;
    const int ml   = lane & 15;
    const float* xb = x + ((size_t)b << 20);   // wave-uniform base
    const int wofs  = ml * Kdim + koff;        // 32-bit element offsets below
    const int ebase = wv * 144 + ml * 9;

    int kp0 = koff,     cofs0 = 0;             // cofs = c<<14
    int kp1 = koff + 1, cofs1 = 0;

    v8f acc0 = v8f{}, acc1 = v8f{};
    for (int kk = 0; kk < 144; ++kk) {
        int e0 = ebase + kp0, e1 = ebase + kp1;
        float a0 = s_msk[e0] * xb[cofs0 + s_idx[e0]];
        float a1 = s_msk[e1] * xb[cofs1 + s_idx[e1]];
        v2f A; A.x = a0; A.y = a1;
        int wk = wofs + (kk << 2);
        v2f B0 = *reinterpret_cast<const v2f*>(wpad + wk);
        v2f B1 = *reinterpret_cast<const v2f*>(wpad + wk + 16 * Kdim);
        acc0 = wmma4(A, B0, acc0);
        acc1 = wmma4(A, B1, acc1);
        kp0 += 4; if (kp0 >= 9) { kp0 -= 9; cofs0 += HW; }
        kp1 += 4; if (kp1 >= 9) { kp1 -= 9; cofs1 += HW; }
    }

    // store N-tile 0: channels 0..15
    {
        float bv = boff[ml];
        int base = ((b * 18 + ml) << 14) + (h0 << 7) + w0 + (hl << 3);
#pragma unroll
        for (int rq = 0; rq < 2; ++rq) {
            float4 vv;
            vv.x = acc0[rq * 4 + 0] + bv;
            vv.y = acc0[rq * 4 + 1] + bv;
            vv.z = acc0[rq * 4 + 2] + bv;
            vv.w = acc0[rq * 4 + 3] + bv;
            *reinterpret_cast<float4*>(off + base + rq * 4) = vv;
        }
    }
    // store N-tile 1: channels 16,17 only
    if (ml < 2) {
        int o = 16 + ml;
        float bv = boff[o];
        int base = ((b * 18 + o) << 14) + (h0 << 7) + w0 + (hl << 3);
#pragma unroll
        for (int rq = 0; rq < 2; ++rq) {
            float4 vv;
            vv.x = acc1[rq * 4 + 0] + bv;
            vv.y = acc1[rq * 4 + 1] + bv;
            vv.z = acc1[rq * 4 + 2] + bv;
            vv.w = acc1[rq * 4 + 3] + bv;
            *reinterpret_cast<float4*>(off + base + rq * 4) = vv;
        }
    }
}

// ---------------------------------------------------------------------------
// Deformable sample + GEMM (M=16 px / wave, N=64, K=576) via WMMA f32.
// ---------------------------------------------------------------------------
template <bool HAS_RES>
__global__ __launch_bounds__(128)
void dgemm_kernel(const float* __restrict__ x,
                  const float* __restrict__ off,
                  const float* __restrict__ wmat,   // (64, 576)
                  const float* __restrict__ bias,
                  const float* __restrict__ res,
                  float* __restrict__ out)
{
    __shared__ __align__(16) int   s_idx[4 * 144 * 4];
    __shared__ __align__(16) float s_wt [4 * 144 * 4];

    const int wv   = threadIdx.x >> 5;
    const int lane = threadIdx.x & 31;
    const int tile = blockIdx.x * 4 + wv;
    const int p0 = tile << 4;
    const int b  = p0 >> 14;
    const int hw = p0 & (HW - 1);
    const int h0 = hw >> 7;
    const int w0 = hw & (Wc - 1);              // multiple of 16

    // ---- per-(pixel,kpos) bilinear descriptors into LDS --------------------
    const float* offp = off + (((size_t)b * 18) << 14) + (h0 << 7);
    for (int ent = lane; ent < 144; ent += 32) {
        int i  = ent / 9;
        int kp = ent - i * 9;
        int wi = w0 + i;
        float dy = offp[((2 * kp)     << 14) + wi];
        float dx = offp[((2 * kp + 1) << 14) + wi];
        int ky = kp / 3;
        int kx = kp - ky * 3;
        float py = (float)(h0 + ky - 1) + dy;
        float px = (float)(wi + kx - 1) + dx;
        float y0f = floorf(py), x0f = floorf(px);
        int   y0  = (int)y0f,   x0i = (int)x0f;
        float fy = py - y0f, fx = px - x0f;
        float gy = 1.f - fy, gx = 1.f - fx;
        int e = ((wv * 144) + ent) << 2;
        int   ys[2] = { y0, y0 + 1 };
        int   xs[2] = { x0i, x0i + 1 };
        float wy[2] = { gy, fy };
        float wx[2] = { gx, fx };
#pragma unroll
        for (int cy = 0; cy < 2; ++cy)
#pragma unroll
            for (int cx = 0; cx < 2; ++cx) {
                int yy = ys[cy], xx = xs[cx];
                bool valid = (yy >= 0) && (yy < Hc) && (xx >= 0) && (xx < Wc);
                int idx = iclamp(yy, 0, Hc - 1) * Wc + iclamp(xx, 0, Wc - 1);
                int slot = e + cy * 2 + cx;
                s_idx[slot] = idx;
                s_wt [slot] = valid ? wy[cy] * wx[cx] : 0.f;
            }
    }
    __syncthreads();

    // ---- main K loop: 32-bit offsets vs wave-uniform bases -----------------
    const int hl   = lane >> 4;
    const int koff = hl << 1;
    const int ml   = lane & 15;
    const float* xb = x + ((size_t)b << 20);   // wave-uniform
    const int wofs  = ml * Kdim + koff;
    const int ebase = (wv * 144 + ml * 9) << 2;

    int kp0 = koff,     cofs0 = 0;             // cofs = c<<14
    int kp1 = koff + 1, cofs1 = 0;

    v8f acc[4] = {v8f{}, v8f{}, v8f{}, v8f{}};

    for (int kk = 0; kk < 144; ++kk) {
        float a[2];
        {
            int e = ebase + (kp0 << 2);
            float v;
            v = s_wt[e + 0] * xb[cofs0 + s_idx[e + 0]];
            v = fmaf(s_wt[e + 1], xb[cofs0 + s_idx[e + 1]], v);
            v = fmaf(s_wt[e + 2], xb[cofs0 + s_idx[e + 2]], v);
            v = fmaf(s_wt[e + 3], xb[cofs0 + s_idx[e + 3]], v);
            a[0] = v;
        }
        {
            int e = ebase + (kp1 << 2);
            float v;
            v = s_wt[e + 0] * xb[cofs1 + s_idx[e + 0]];
            v = fmaf(s_wt[e + 1], xb[cofs1 + s_idx[e + 1]], v);
            v = fmaf(s_wt[e + 2], xb[cofs1 + s_idx[e + 2]], v);
            v = fmaf(s_wt[e + 3], xb[cofs1 + s_idx[e + 3]], v);
            a[1] = v;
        }
        v2f A; A.x = a[0]; A.y = a[1];
        int wk = wofs + (kk << 2);
#pragma unroll
        for (int n = 0; n < 4; ++n) {
            v2f Bv = *reinterpret_cast<const v2f*>(wmat + wk + n * 16 * Kdim);
            acc[n] = wmma4(A, Bv, acc[n]);
        }
        kp0 += 4; if (kp0 >= 9) { kp0 -= 9; cofs0 += HW; }
        kp1 += 4; if (kp1 >= 9) { kp1 -= 9; cofs1 += HW; }
    }

    // ---- epilogue ----------------------------------------------------------
#pragma unroll
    for (int n = 0; n < 4; ++n) {
        int o = (n << 4) + ml;
        float bv = bias[o];
        int base = ((b * Cc + o) << 14) + (h0 << 7) + w0 + (hl << 3);
#pragma unroll
        for (int rq = 0; rq < 2; ++rq) {
            float4 vv;
            vv.x = acc[n][rq * 4 + 0] + bv;
            vv.y = acc[n][rq * 4 + 1] + bv;
            vv.z = acc[n][rq * 4 + 2] + bv;
            vv.w = acc[n][rq * 4 + 3] + bv;
            if constexpr (HAS_RES) {
                float4 rr = *reinterpret_cast<const float4*>(res + base + rq * 4);
                vv.x += rr.x; vv.y += rr.y; vv.z += rr.z; vv.w += rr.w;
            }
            vv.x = vv.x > 0.f ? vv.x : NEGS * vv.x;
            vv.y = vv.y > 0.f ? vv.y : NEGS * vv.y;
            vv.z = vv.z > 0.f ? vv.z : NEGS * vv.z;
            vv.w = vv.w > 0.f ? vv.w : NEGS * vv.w;
            *reinterpret_cast<float4*>(out + base + rq * 4) = vv;
        }
    }
}

// ---------------------------------------------------------------------------
extern "C" void kernel_launch(void* const* d_in, const int* in_sizes, int n_in,
                              void* d_out, int out_size, void* d_ws, size_t ws_size,
                              hipStream_t stream)
{
    const float* x     = (const float*)d_in[0];
    const float* woff1 = (const float*)d_in[1];
    const float* boff1 = (const float*)d_in[2];
    const float* w1    = (const float*)d_in[3];
    const float* b1    = (const float*)d_in[4];
    const float* woff2 = (const float*)d_in[5];
    const float* boff2 = (const float*)d_in[6];
    const float* w2    = (const float*)d_in[7];
    const float* b2    = (const float*)d_in[8];
    float* out = (float*)d_out;

    // workspace layout (floats): off | hbuf | wpad
    float* off  = (float*)d_ws;
    float* hbuf = off  + (size_t)Bc * 18 * HW;
    float* wpad = hbuf + (size_t)Bc * Cc * HW;

    dim3 gP((32 * Kdim + 255) / 256), bP(256);
    dim3 gG((Bc * HW / 16) / 4),      bG(128);

    // layer 1
    padw_kernel        <<<gP, bP, 0, stream>>>(woff1, wpad);
    offconv_kernel     <<<gG, bG, 0, stream>>>(x, wpad, boff1, off);
    dgemm_kernel<false><<<gG, bG, 0, stream>>>(x, off, w1, b1, nullptr, hbuf);
    // layer 2
    padw_kernel        <<<gP, bP, 0, stream>>>(woff2, wpad);
    offconv_kernel     <<<gG, bG, 0, stream>>>(hbuf, wpad, boff2, off);
    dgemm_kernel<true> <<<gG, bG, 0, stream>>>(hbuf, off, w2, b2, x, out);
}